// MoEExperts_9517647528140
// MI455X (gfx1250) — compile-verified
//
#include <hip/hip_runtime.h>
#include <hip/hip_bf16.h>

typedef __attribute__((ext_vector_type(2))) float v2f;
typedef __attribute__((ext_vector_type(8))) float v8f;

#define TT   2048          // tokens
#define KK   2             // top-k
#define EE   8             // experts
#define FF   2048          // ffn hidden
#define HH   768           // hidden
#define CAP  4096          // per-expert list capacity (worst case all slots)
#define MT   32            // token-tile (rows per block tile)
#define GROWS (TT*KK + EE*MT)   // slot rows + per-expert pad (dump) rows

static __device__ __forceinline__ v8f wmma4(v2f a, v2f b, v8f c) {
    // V_WMMA_F32_16X16X4_F32: D = A(16x4) * B(4x16) + C(16x16), wave32
    return __builtin_amdgcn_wmma_f32_16x16x4_f32(
        /*neg_a=*/false, a, /*neg_b=*/false, b,
        /*c_mod=*/(short)0, c, /*reuse_a=*/false, /*reuse_b=*/false);
}

// ---------------------------------------------------------------------------
// Kernel 1: routing. Build per-expert (token, slot-id) lists, pad to MT.
// ---------------------------------------------------------------------------
__global__ __launch_bounds__(256) void route_kernel(
    const int* __restrict__ te, int* __restrict__ ltok,
    int* __restrict__ lsid, int* __restrict__ tiles)
{
    __shared__ int cnt[EE];
    int tid = threadIdx.x;
    if (tid < EE) cnt[tid] = 0;
    __syncthreads();
    for (int s = tid; s < TT * KK; s += 256) {
        int e = te[s];
        int p = atomicAdd(&cnt[e], 1);
        ltok[e * CAP + p] = s >> 1;   // token = slot / K
        lsid[e * CAP + p] = s;        // global slot id
    }
    __syncthreads();
    if (tid < EE) {
        int c = cnt[tid];
        int padded = (c + MT - 1) & ~(MT - 1);
        for (int i = c; i < padded; ++i) {   // pad with token 0 -> dump rows
            ltok[tid * CAP + i] = 0;
            lsid[tid * CAP + i] = TT * KK + tid * MT + (i - c);
        }
        tiles[tid] = padded / MT;
    }
}

// ---------------------------------------------------------------------------
// Kernel 2: GEMM1 (x @ W1^T, x @ V1^T) + exact GELU gate -> G[slot][F]
// grid: (tile, expert, fchunk of 256); block: 256 = 8 waves.
// wave region: 32 slots x 32 F cols, both W1 and V1 -> 8 wmma / k-step
// fed by 4 b64 B loads (B reused across 2 M-tiles) + 2 LDS A loads.
// ---------------------------------------------------------------------------
__global__ __launch_bounds__(256) void expert_gemm1_kernel(
    const float* __restrict__ x, const float* __restrict__ w1,
    const float* __restrict__ v1, const int* __restrict__ tiles,
    const int* __restrict__ ltok, const int* __restrict__ lsid,
    float* __restrict__ G)
{
    int e = blockIdx.y;
    int tile = blockIdx.x;
    if (tile >= tiles[e]) return;

    __shared__ float xs[MT * 772];   // 32 gathered token rows, stride 772 (bank-safe)
    __shared__ int stok[MT], ssid[MT];
    int tid = threadIdx.x;
    if (tid < MT) {
        stok[tid] = ltok[e * CAP + tile * MT + tid];
        ssid[tid] = lsid[e * CAP + tile * MT + tid];
    }
    __syncthreads();
    for (int idx = tid; idx < MT * (HH / 4); idx += 256) {
        int row = idx / (HH / 4);
        int q = idx - row * (HH / 4);
        ((float4*)(xs + row * 772))[q] =
            ((const float4*)(x + (size_t)stok[row] * HH))[q];
    }
    __syncthreads();

    int wave = tid >> 5, lane = tid & 31;
    int m = lane & 15, half = lane >> 4;
    int fbase = (int)blockIdx.z * 256 + wave * 32;

    const float* ap0 = xs + m * 772 + half * 2;
    const float* ap1 = xs + (m + 16) * 772 + half * 2;
    const float* b1p[2];
    const float* b2p[2];
#pragma unroll
    for (int j = 0; j < 2; ++j) {
        size_t r = (size_t)(e * FF + fbase + j * 16 + m) * HH + half * 2;
        b1p[j] = w1 + r;
        b2p[j] = v1 + r;
    }

    v8f h1[2][2] = {};   // [m-tile][n-tile]
    v8f h2[2][2] = {};
#pragma unroll 2
    for (int k = 0; k < HH; k += 4) {
        v2f a0 = *(const v2f*)(ap0 + k);
        v2f a1 = *(const v2f*)(ap1 + k);
#pragma unroll
        for (int j = 0; j < 2; ++j) {
            v2f b1 = *(const v2f*)(b1p[j] + k);
            h1[0][j] = wmma4(a0, b1, h1[0][j]);
            h1[1][j] = wmma4(a1, b1, h1[1][j]);
            v2f b2 = *(const v2f*)(b2p[j] + k);
            h2[0][j] = wmma4(a0, b2, h2[0][j]);
            h2[1][j] = wmma4(a1, b2, h2[1][j]);
        }
    }

#pragma unroll
    for (int mt = 0; mt < 2; ++mt) {
#pragma unroll
        for (int j = 0; j < 2; ++j) {
#pragma unroll
            for (int r = 0; r < 8; ++r) {
                int row = mt * 16 + r + half * 8;
                float a1 = h1[mt][j][r];
                float a2 = h2[mt][j][r];
                float g = 0.5f * a1 * (1.0f + erff(a1 * 0.70710678118654752f)) * a2;
                G[(size_t)ssid[row] * FF + (fbase + j * 16 + m)] = g;
            }
        }
    }
}

// ---------------------------------------------------------------------------
// Kernel 3: GEMM2: slot_out[slot] = G[slot] @ W2[e]   (32 slots x 768 per block)
// grid: (tile, expert); block: 384 = 12 waves; wave owns 32x64 of H.
// 8 wmma / k-step fed by 8 coalesced b32 B loads (reused across 2 M-tiles).
// ---------------------------------------------------------------------------
__global__ __launch_bounds__(384) void expert_gemm2_kernel(
    const float* __restrict__ w2, const int* __restrict__ tiles,
    const int* __restrict__ lsid, const float* __restrict__ G,
    float* __restrict__ so)
{
    int e = blockIdx.y;
    int tile = blockIdx.x;
    if (tile >= tiles[e]) return;

    __shared__ int ssid[MT];
    int tid = threadIdx.x;
    if (tid < MT) ssid[tid] = lsid[e * CAP + tile * MT + tid];
    __syncthreads();

    int wave = tid >> 5, lane = tid & 31;
    int m = lane & 15, half = lane >> 4;
    int h0 = wave * 64;

    const float* ap0 = G + (size_t)ssid[m] * FF + half * 2;       // gathered G rows
    const float* ap1 = G + (size_t)ssid[m + 16] * FF + half * 2;
    const float* bbase = w2 + (size_t)(e * FF) * HH + h0 + m;     // coalesced W2 cols

    v8f acc[2][4] = {};
#pragma unroll 2
    for (int k = 0; k < FF; k += 4) {
        v2f a0 = *(const v2f*)(ap0 + k);
        v2f a1 = *(const v2f*)(ap1 + k);
        const float* b0 = bbase + (size_t)(k + half * 2) * HH;
#pragma unroll
        for (int j = 0; j < 4; ++j) {
            v2f b;
            b.x = b0[j * 16];
            b.y = b0[j * 16 + HH];
            acc[0][j] = wmma4(a0, b, acc[0][j]);
            acc[1][j] = wmma4(a1, b, acc[1][j]);
        }
    }

#pragma unroll
    for (int mt = 0; mt < 2; ++mt) {
#pragma unroll
        for (int j = 0; j < 4; ++j) {
#pragma unroll
            for (int r = 0; r < 8; ++r) {
                int row = mt * 16 + r + half * 8;
                so[(size_t)ssid[row] * HH + (h0 + j * 16 + m)] = acc[mt][j][r];
            }
        }
    }
}

// ---------------------------------------------------------------------------
// Kernel 4: combine the two routed slots per token with their gate weights.
// ---------------------------------------------------------------------------
__global__ __launch_bounds__(256) void combine_kernel(
    const float* __restrict__ tw, const float* __restrict__ so,
    float* __restrict__ out)
{
    int idx = blockIdx.x * 256 + threadIdx.x;      // float4 index over [T, H]
    int t = idx / (HH / 4);
    int q = idx - t * (HH / 4);
    float wA = tw[2 * t];
    float wB = tw[2 * t + 1];
    float4 a = ((const float4*)(so + (size_t)(2 * t) * HH))[q];
    float4 b = ((const float4*)(so + (size_t)(2 * t + 1) * HH))[q];
    float4 o;
    o.x = wA * a.x + wB * b.x;
    o.y = wA * a.y + wB * b.y;
    o.z = wA * a.z + wB * b.z;
    o.w = wA * a.w + wB * b.w;
    ((float4*)out)[idx] = o;
}

// ---------------------------------------------------------------------------
extern "C" void kernel_launch(void* const* d_in, const int* in_sizes, int n_in,
                              void* d_out, int out_size, void* d_ws, size_t ws_size,
                              hipStream_t stream) {
    const float* x  = (const float*)d_in[0];   // [1, 2048, 768]
    const float* tw = (const float*)d_in[1];   // [2048, 2]
    const float* w1 = (const float*)d_in[2];   // [E*F, H]
    const float* v1 = (const float*)d_in[3];
    const float* w2 = (const float*)d_in[4];
    const int*   te = (const int*)d_in[5];     // [2048, 2]

    char* ws = (char*)d_ws;
    int* tiles = (int*)ws;                                   // E ints (256B reserved)
    int* ltok  = (int*)(ws + 256);                           // E*CAP ints
    int* lsid  = ltok + EE * CAP;                            // E*CAP ints
    float* G   = (float*)(ws + 256 + 2 * (size_t)EE * CAP * sizeof(int)); // GROWS*FF
    float* so  = G + (size_t)GROWS * FF;                     // GROWS*HH
    float* out = (float*)d_out;

    route_kernel<<<1, 256, 0, stream>>>(te, ltok, lsid, tiles);
    expert_gemm1_kernel<<<dim3(128, EE, 8), 256, 0, stream>>>(
        x, w1, v1, tiles, ltok, lsid, G);
    expert_gemm2_kernel<<<dim3(128, EE), 384, 0, stream>>>(
        w2, tiles, lsid, G, so);
    combine_kernel<<<(TT * HH / 4) / 256, 256, 0, stream>>>(tw, so, out);
}